// Propagation_67430986547242
// MI455X (gfx1250) — compile-verified
//
#include <hip/hip_runtime.h>

// Dilated 3x3 unfold (reflect pad) for two [2,64,256,256] fp32 tensors.
// CDNA5 formulation: async global->LDS staging (rows fetched once per band),
// reflected halo built in LDS, then 9 shifted copies streamed LDS->HBM via
// global_store_async_from_lds + TH_STORE_NT (no VGPR round-trip, ASYNCcnt
// tracked). Store-bandwidth-bound (~604 MB written once @ 23.3 TB/s ~ 26us);
// interior bands use affine strength-reduced addressing (~3 VALU per async
// store) so instruction issue stays well under the HBM store roofline.

typedef float v4f __attribute__((ext_vector_type(4)));
typedef __attribute__((address_space(3))) float lds_float_t;

#define B_ 2
#define C_ 64
#define H_ 256
#define W_ 256
#define K_ 9
#define ROWS_ 16                       // output rows per block
#define DMAX_ 8                        // max dilation on the LDS fast path
#define HALO_ DMAX_
#define STRIDE_ (W_ + 2 * DMAX_)       // 272 floats per LDS row
#define LROWS_ (ROWS_ + 2 * DMAX_)     // 32 staged rows max

__device__ __forceinline__ int reflect_idx(int i, int n) {
    // jnp.pad mode="reflect" (no edge repeat) -> branchless v_cndmask
    i = (i < 0) ? -i : i;
    i = (i >= n) ? (2 * n - 2 - i) : i;
    return i;
}

__device__ __forceinline__ unsigned lds_byte_addr(float* p) {
    // generic -> LDS addrspace cast yields the 32-bit LDS offset the async
    // engine expects (do not assume the shared array sits at offset 0).
    return (unsigned)(unsigned long long)(lds_float_t*)p;
}

__device__ __forceinline__ void async_load_lds_b128(unsigned lds_b, const float* g) {
    asm volatile("global_load_async_to_lds_b128 %0, %1, off"
                 :: "v"(lds_b), "v"(g) : "memory");
}
__device__ __forceinline__ void async_store_lds_b64(const float* g, unsigned lds_b) {
    asm volatile("global_store_async_from_lds_b64 %0, %1, off th:TH_STORE_NT"
                 :: "v"(g), "v"(lds_b) : "memory");
}
__device__ __forceinline__ void async_store_lds_b32(const float* g, unsigned lds_b) {
    asm volatile("global_store_async_from_lds_b32 %0, %1, off th:TH_STORE_NT"
                 :: "v"(g), "v"(lds_b) : "memory");
}
__device__ __forceinline__ void wait_async0() {
    asm volatile("s_wait_asynccnt 0x0" ::: "memory");
}

__global__ __launch_bounds__(256) void
unfold_dilated_async_kernel(const float* __restrict__ x,
                            const float* __restrict__ y,
                            float* __restrict__ out,
                            const int* __restrict__ dil)
{
    __shared__ __align__(16) float tile[LROWS_ * STRIDE_];

    const int d   = __builtin_amdgcn_readfirstlane(dil[0]);
    const int tid = threadIdx.x;
    const int h0  = blockIdx.x * ROWS_;
    const int c   = blockIdx.y;            // channel
    const int b   = blockIdx.z & 1;        // batch
    const int t   = blockIdx.z >> 1;       // tensor: 0=x, 1=y

    const float* __restrict__ src = (t == 0) ? x : y;
    const float* __restrict__ splane =
        src + (((long long)b * C_ + c) * H_) * (long long)W_;
    // k=0 output plane; plane for k is obase + k*C*H*W (channel = k*C + c)
    float* __restrict__ obase =
        out + ((((long long)(t * B_ + b)) * K_ * C_ + c) * H_) * (long long)W_;

    if (d <= DMAX_) {
        const int rlo   = (h0 - d > 0) ? (h0 - d) : 0;
        const int rhi_  = h0 + ROWS_ - 1 + d;
        const int rhi   = (rhi_ < H_ - 1) ? rhi_ : (H_ - 1);
        const int nrows = rhi - rlo + 1;
        const unsigned tbase = lds_byte_addr(tile);

        // ---- Phase 1: async global -> LDS (b128), one fetch per input row.
        const int chunks = nrows * (W_ / 4);
        for (int q0 = 0; q0 < chunks; q0 += 256) {
            int q = q0 + tid;
            q = (q < chunks) ? q : (chunks - 1);   // full EXEC; dup writes same data
            const int r  = q >> 6;                 // W_/4 == 64
            const int wc = (q & 63) << 2;
            const unsigned lb = tbase + (unsigned)((r * STRIDE_ + HALO_ + wc) * 4);
            async_load_lds_b128(lb, splane + (long long)(rlo + r) * W_ + wc);
        }
        wait_async0();
        __syncthreads();

        // ---- Phase 2: build horizontal reflected halo in LDS (DS ops).
        for (int i = tid; i < nrows * 2 * DMAX_; i += 256) {
            const int r    = i / (2 * DMAX_);
            const int rem  = i - r * 2 * DMAX_;
            const int side = rem >> 3;             // DMAX_ == 8
            const int j    = (rem & (DMAX_ - 1)) + 1;
            if (j <= d) {
                float* row = tile + r * STRIDE_ + HALO_;
                if (side == 0) row[-j]         = row[j];
                else           row[W_ - 1 + j] = row[W_ - 1 - j];
            }
        }
        __syncthreads();

        // ---- Phase 3: 9 shifted copies streamed LDS -> HBM (async, NT).
        // +/-d float shift => pick granule from d's parity for alignment.
        const bool interior = (h0 >= d) && (h0 + ROWS_ - 1 + d <= H_ - 1);

        if (interior && (d & 1) == 0) {
            // Fast path: no vertical reflection (lr = row + voff + d) and all
            // addresses affine in the pass index -> 2 increments per store.
            // Per pass: 256 lanes x 8B = 2 rows. Global float offset
            //   = h0*W + 512*pass + 2*tid ; LDS lane part loop-invariant.
            const unsigned lane_lds =
                (unsigned)((((tid >> 7) * STRIDE_) + ((tid & 127) << 1)) * 4);
            float* __restrict__ grow0 = obase + (long long)h0 * W_ + (tid << 1);
#pragma unroll
            for (int k = 0; k < K_; ++k) {
                const int ki   = k / 3;
                const int kj   = k - 3 * ki;
                const int voff = (ki - 1) * d;
                const int hoff = (kj - 1) * d;
                float* g = grow0 + (long long)k * (C_ * H_ * (long long)W_);
                unsigned lb = tbase + lane_lds +
                    (unsigned)((((voff + d) * STRIDE_) + HALO_ + hoff) * 4);
#pragma unroll
                for (int pass = 0; pass < 8; ++pass) {
                    async_store_lds_b64(g, lb);
                    g  += 2 * W_;                    // 512 floats
                    lb += 2 * STRIDE_ * 4;           // 2176 bytes
                }
            }
        } else {
#pragma unroll
            for (int k = 0; k < K_; ++k) {
                const int ki   = k / 3;
                const int kj   = k - 3 * ki;
                const int voff = (ki - 1) * d;
                const int hoff = (kj - 1) * d;
                float* __restrict__ okp =
                    obase + (long long)k * (C_ * H_ * (long long)W_);

                if ((d & 1) == 0) {           // uniform branch; b64 granule
                    for (int q0 = 0; q0 < ROWS_ * (W_ / 2); q0 += 256) { // 2048
                        const int q   = q0 + tid;
                        const int row = q >> 7;
                        const int w   = (q & 127) << 1;
                        const int lr  = reflect_idx(h0 + row + voff, H_) - rlo;
                        const unsigned lb = tbase +
                            (unsigned)((lr * STRIDE_ + HALO_ + w + hoff) * 4);
                        async_store_lds_b64(okp + (long long)(h0 + row) * W_ + w, lb);
                    }
                } else {                      // odd d: b32 granule
                    for (int q0 = 0; q0 < ROWS_ * W_; q0 += 256) {       // 4096
                        const int q   = q0 + tid;
                        const int row = q >> 8;
                        const int w   = q & 255;
                        const int lr  = reflect_idx(h0 + row + voff, H_) - rlo;
                        const unsigned lb = tbase +
                            (unsigned)((lr * STRIDE_ + HALO_ + w + hoff) * 4);
                        async_store_lds_b32(okp + (long long)(h0 + row) * W_ + w, lb);
                    }
                }
            }
        }
        // S_ENDPGM implicit wait-idle drains outstanding async stores.
    } else {
        // ---- Generic fallback (any d): direct gather + NT b128 store.
        for (int q0 = 0; q0 < K_ * ROWS_ * (W_ / 4); q0 += 256) {        // 9216
            const int q   = q0 + tid;
            const int w4  = q & 63;
            const int row = (q >> 6) & (ROWS_ - 1);
            const int k   = q >> 10;
            const int ki  = k / 3;
            const int kj  = k - 3 * ki;
            const int sh  = reflect_idx(h0 + row + (ki - 1) * d, H_);
            const float* __restrict__ srow = splane + (long long)sh * W_;
            const int w  = w4 << 2;
            const int s0 = w + (kj - 1) * d;
            v4f v;
            if (s0 >= 0 && s0 + 3 < W_) {
                v.x = srow[s0]; v.y = srow[s0 + 1];
                v.z = srow[s0 + 2]; v.w = srow[s0 + 3];
            } else {
                v.x = srow[reflect_idx(s0,     W_)];
                v.y = srow[reflect_idx(s0 + 1, W_)];
                v.z = srow[reflect_idx(s0 + 2, W_)];
                v.w = srow[reflect_idx(s0 + 3, W_)];
            }
            float* op = obase + (long long)k * (C_ * H_ * (long long)W_)
                              + (long long)(h0 + row) * W_ + w;
            __builtin_nontemporal_store(v, (v4f*)op);
        }
    }
}

extern "C" void kernel_launch(void* const* d_in, const int* in_sizes, int n_in,
                              void* d_out, int out_size, void* d_ws, size_t ws_size,
                              hipStream_t stream) {
    (void)in_sizes; (void)n_in; (void)out_size; (void)d_ws; (void)ws_size;

    const float* x   = (const float*)d_in[0];
    const float* y   = (const float*)d_in[1];
    const int*   dil = (const int*)d_in[2];
    float*       out = (float*)d_out;

    dim3 grid(H_ / ROWS_,   // 16 row-bands
              C_,           // 64 channels
              2 * B_);      // (tensor, batch)
    dim3 block(256);        // 8 wave32s

    unfold_dilated_async_kernel<<<grid, block, 0, stream>>>(x, y, out, dil);
}